// Router_12051678232640
// MI455X (gfx1250) — compile-verified
//
#include <hip/hip_runtime.h>

typedef __attribute__((ext_vector_type(2))) float v2f;
typedef __attribute__((ext_vector_type(8))) float v8f;

#define NEXP 64
#define CDIM 1024          // feature dim (fixed by the reference)

// ---------------------------------------------------------------------------
// Kernel 1: fused router GEMM (V_WMMA_F32_16X16X4_F32) + top-2 + softmax(2)
// 1024-thread WG (32 waves) shares one 256 KB LDS staging of w_g -> full WGP
// occupancy (8 waves/SIMD) hides global A-stream latency. K loop is
// double-buffered for both A (global b64) and B (ds b64) fragments.
// ---------------------------------------------------------------------------
__global__ void router_gemm_topk(const float* __restrict__ x,
                                 const float* __restrict__ wg,
                                 int* __restrict__ eSel,
                                 float* __restrict__ pSel,
                                 float* __restrict__ outIdx,   // d_out + 130*N
                                 int N) {
    extern __shared__ float ldsB[];       // 64 * 1024 floats = 256 KB

    const int tid  = threadIdx.x;         // 1024 threads = 32 waves
    const int lane = tid & 31;
    const int wv   = tid >> 5;
    const int m    = lane & 15;           // token row / expert column within tile
    const int kh   = (lane >> 4) << 1;    // 0 or 2: K sub-offset per half-wave

    // ---- stage w_g into LDS with gfx1250 async copy (16384 x b128) ----
    {
        const float4* wg4 = (const float4*)wg;
        for (int i = tid; i < 16 * CDIM; i += 1024) {
            unsigned ldsOff = (unsigned)(uintptr_t)(&ldsB[i * 4]);
            const float4* gp = wg4 + i;
            asm volatile("global_load_async_to_lds_b128 %0, %1, off"
                         :: "v"(ldsOff), "v"(gp) : "memory");
        }
        asm volatile("s_wait_asynccnt 0x0" ::: "memory");
        __syncthreads();
    }

    // per-lane B fragment base offsets (element index into ldsB)
    const int bo0 = (0 * 16 + m) * CDIM + kh;
    const int bo1 = (1 * 16 + m) * CDIM + kh;
    const int bo2 = (2 * 16 + m) * CDIM + kh;
    const int bo3 = (3 * 16 + m) * CDIM + kh;

    const int tile    = blockIdx.x * 32 + wv;   // one 16-token tile per wave
    const int tokBase = tile * 16;
    const float* xa = x + (size_t)(tokBase + m) * CDIM + kh;

    v8f zero = {0.f,0.f,0.f,0.f,0.f,0.f,0.f,0.f};
    v8f acc[4] = {zero, zero, zero, zero};

    // ---- software-pipelined K loop (double-buffer A and B fragments) ----
    v2f aCur = *(const v2f*)(xa);
    v2f b0c  = *(const v2f*)&ldsB[bo0];
    v2f b1c  = *(const v2f*)&ldsB[bo1];
    v2f b2c  = *(const v2f*)&ldsB[bo2];
    v2f b3c  = *(const v2f*)&ldsB[bo3];

    for (int k0 = 0; k0 < CDIM - 4; k0 += 4) {
        v2f aN  = *(const v2f*)(xa + k0 + 4);
        v2f b0n = *(const v2f*)&ldsB[bo0 + k0 + 4];
        v2f b1n = *(const v2f*)&ldsB[bo1 + k0 + 4];
        v2f b2n = *(const v2f*)&ldsB[bo2 + k0 + 4];
        v2f b3n = *(const v2f*)&ldsB[bo3 + k0 + 4];
        acc[0] = __builtin_amdgcn_wmma_f32_16x16x4_f32(false, aCur, false, b0c, (short)0, acc[0], false, false);
        acc[1] = __builtin_amdgcn_wmma_f32_16x16x4_f32(false, aCur, false, b1c, (short)0, acc[1], false, false);
        acc[2] = __builtin_amdgcn_wmma_f32_16x16x4_f32(false, aCur, false, b2c, (short)0, acc[2], false, false);
        acc[3] = __builtin_amdgcn_wmma_f32_16x16x4_f32(false, aCur, false, b3c, (short)0, acc[3], false, false);
        aCur = aN; b0c = b0n; b1c = b1n; b2c = b2n; b3c = b3n;
    }
    acc[0] = __builtin_amdgcn_wmma_f32_16x16x4_f32(false, aCur, false, b0c, (short)0, acc[0], false, false);
    acc[1] = __builtin_amdgcn_wmma_f32_16x16x4_f32(false, aCur, false, b1c, (short)0, acc[1], false, false);
    acc[2] = __builtin_amdgcn_wmma_f32_16x16x4_f32(false, aCur, false, b2c, (short)0, acc[2], false, false);
    acc[3] = __builtin_amdgcn_wmma_f32_16x16x4_f32(false, aCur, false, b3c, (short)0, acc[3], false, false);

    // C layout: VGPR r holds token M = r (lanes 0-15) / r+8 (lanes 16-31),
    // expert = 16*t + (lane&15) for accumulator t. Top-2 per token:
#pragma unroll
    for (int r = 0; r < 8; ++r) {
        float v1 = -3.4e38f, v2c = -3.4e38f;
        int   i1 = 0,        i2  = 0;
#pragma unroll
        for (int t = 0; t < 4; ++t) {
            float v = acc[t][r];
            int   e = t * 16 + m;   // ascending e: strict '>' keeps lowest index on ties
            if (v > v1)       { v2c = v1; i2 = i1; v1 = v; i1 = e; }
            else if (v > v2c) { v2c = v;  i2 = e; }
        }
        // butterfly over the 16-lane half (xor 8,4,2,1 never crosses lane 16)
#pragma unroll
        for (int off = 8; off >= 1; off >>= 1) {
            float w1 = __shfl_xor(v1,  off, 32);
            int   j1 = __shfl_xor(i1,  off, 32);
            float w2 = __shfl_xor(v2c, off, 32);
            int   j2 = __shfl_xor(i2,  off, 32);
            bool aF = (v1 > w1) || (v1 == w1 && i1 < j1);
            float h  = aF ? v1 : w1;  int hi = aF ? i1 : j1;
            float sx = aF ? v2c : w2; int si = aF ? i2 : j2;   // winner's second
            float sy = aF ? w1 : v1;  int sj = aF ? j1 : i1;   // loser's head
            bool sF = (sx > sy) || (sx == sy && si < sj);
            v1 = h; i1 = hi;
            v2c = sF ? sx : sy; i2 = sF ? si : sj;
        }
        if (m == 0) {
            int n = tokBase + r + ((lane >> 4) << 3);
            float p0 = 1.0f / (1.0f + __expf(v2c - v1));
            float p1 = 1.0f / (1.0f + __expf(v1 - v2c));
            eSel[n]     = i1;  eSel[N + n] = i2;   // item order j = k*N + n
            pSel[n]     = p0;  pSel[N + n] = p1;
            outIdx[2 * n]     = (float)i1;
            outIdx[2 * n + 1] = (float)i2;
        }
    }
}

// ---------------------------------------------------------------------------
// Kernel 2: coalesced zero-fill of the one-hot mask region [N*2*64 floats]
// ---------------------------------------------------------------------------
__global__ void zero_mask(float4* __restrict__ p) {
    size_t i = (size_t)blockIdx.x * blockDim.x + threadIdx.x;
    p[i] = make_float4(0.f, 0.f, 0.f, 0.f);
}

// ---------------------------------------------------------------------------
// Kernel 3: per-block expert histogram over the 2N item sequence
// ---------------------------------------------------------------------------
__global__ void router_hist(const int* __restrict__ eSel, int* __restrict__ hist) {
    __shared__ int h[NEXP];
    int tid = threadIdx.x;
    if (tid < NEXP) h[tid] = 0;
    __syncthreads();
    int j = blockIdx.x * blockDim.x + tid;
    atomicAdd(&h[eSel[j]], 1);
    __syncthreads();
    if (tid < NEXP) hist[blockIdx.x * NEXP + tid] = h[tid];
}

// ---------------------------------------------------------------------------
// Kernel 4: exclusive scan over blocks per expert (tiny, one workgroup)
// ---------------------------------------------------------------------------
__global__ void router_scan(const int* __restrict__ hist, int* __restrict__ offs,
                            int numBlk) {
    int e = threadIdx.x;   // 64 threads
    int run = 0;
    for (int b = 0; b < numBlk; ++b) {
        int t = hist[b * NEXP + e];
        offs[b * NEXP + e] = run;
        run += t;
    }
}

// ---------------------------------------------------------------------------
// Kernel 5: deterministic ordered rank + capacity mask + outputs
// ---------------------------------------------------------------------------
__global__ void router_finalize(const int* __restrict__ eSel,
                                const float* __restrict__ pSel,
                                const int* __restrict__ offs,
                                float* __restrict__ out,
                                int N, int capacity) {
    __shared__ int wh[32 * NEXP];           // per-wave expert histograms
    int tid  = threadIdx.x;                 // 1024 threads = 32 waves
    int lane = tid & 31, wv = tid >> 5;
    int j = blockIdx.x * 1024 + tid;
    int   e = eSel[j];
    float p = pSel[j];

    wh[tid] = 0; wh[tid + 1024] = 0;
    __syncthreads();

    // match_any via 6 ballots (experts are 6-bit)
    unsigned mask = 0xffffffffu;
#pragma unroll
    for (int b = 0; b < 6; ++b) {
        unsigned bal = (unsigned)__ballot((e >> b) & 1);
        mask &= ((e >> b) & 1) ? bal : ~bal;
    }
    int laneRank = __popc(mask & ((1u << lane) - 1u));
    if (laneRank == 0) wh[wv * NEXP + e] = __popc(mask);   // group leader
    __syncthreads();

    if (tid < NEXP) {   // exclusive prefix across waves, per expert
        int run = 0;
        for (int w = 0; w < 32; ++w) {
            int t = wh[w * NEXP + tid];
            wh[w * NEXP + tid] = run;
            run += t;
        }
    }
    __syncthreads();

    int rank  = offs[blockIdx.x * NEXP + e] + wh[wv * NEXP + e] + laneRank;
    bool incap = rank < capacity;

    int k = (j >= N) ? 1 : 0;
    int n = j - k * N;
    int pi = n * 2 + k;
    out[(size_t)N * 128 + pi] = incap ? p : 0.0f;   // router_probs_masked
    out[(size_t)N * 132 + pi] = (float)rank;        // final_rank (unmasked by capacity)
    if (incap) out[(size_t)pi * 64 + e] = 1.0f;     // final_expert_mask one-hot
}

// ---------------------------------------------------------------------------
extern "C" void kernel_launch(void* const* d_in, const int* in_sizes, int n_in,
                              void* d_out, int out_size, void* d_ws, size_t ws_size,
                              hipStream_t stream) {
    const float* x  = (const float*)d_in[0];
    const float* wg = (const float*)d_in[1];
    float* out = (float*)d_out;

    int C = in_sizes[1] / NEXP;           // 1024
    int N = in_sizes[0] / C;              // 32768 tokens

    // eval-mode capacity: floor(TOP_K * 2.0 * N / 64), round up to even, min 4
    int capacity = (int)((4.0 * (double)N) / 64.0);
    capacity += capacity & 1;
    if (capacity < 4) capacity = 4;

    char* ws = (char*)d_ws;
    int*   eSel = (int*)ws;                                 // [2N]
    float* pSel = (float*)(ws + (size_t)2 * N * 4);         // [2N]
    int*   hist = (int*)(ws + (size_t)4 * N * 4);           // [numBlk*64]
    int numBlk  = (2 * N) / 1024;                           // 64
    int*   offs = hist + (size_t)numBlk * NEXP;             // [numBlk*64]

    // 1. zero the one-hot mask region (N*128 floats = 32N float4)
    zero_mask<<<(N / 8), 256, 0, stream>>>((float4*)out);

    // 2. WMMA GEMM + top-2 + softmax; also writes top_k_indices at 130*N
    //    grid: (N/16) tiles / 32 waves = 64 blocks of 1024 threads;
    //    256 KB dynamic LDS -> one WG fills a WGP (8 waves/SIMD32)
    router_gemm_topk<<<(N / 16) / 32, 1024, NEXP * CDIM * sizeof(float), stream>>>(
        x, wg, eSel, pSel, out + (size_t)130 * N, N);

    // 3. block histograms over 2N items
    router_hist<<<numBlk, 1024, 0, stream>>>(eSel, hist);

    // 4. exclusive block scan per expert
    router_scan<<<1, NEXP, 0, stream>>>(hist, offs, numBlk);

    // 5. ordered ranks, capacity mask, outputs
    router_finalize<<<numBlk, 1024, 0, stream>>>(eSel, pSel, offs, out, N, capacity);
}